// fMRI3DGNN_16595753631820
// MI455X (gfx1250) — compile-verified
//
#include <hip/hip_runtime.h>
#include <math.h>

typedef float v2f __attribute__((ext_vector_type(2)));
typedef float v8f __attribute__((ext_vector_type(8)));

#define NODES 200
#define NN    40000   // 200*200

// ---------------- workspace layout (bytes) ----------------
static const size_t WS_APAD = 0;          // 16*40000*4      = 2,560,000
static const size_t WS_P    = 2560000;    // 4*40000*4       =   640,000
static const size_t WS_ADJ  = 3200000;    // 4*40000*4       =   640,000
static const size_t WS_BASE = 3840000;    // 800*2*4         =     6,400
static const size_t WS_SEL  = 3846400;    // 8*4
static const size_t WS_THR  = 3846432;    // 4*4
static const size_t WS_X0   = 3846448;    // 800*16*4        =    51,200
static const size_t WS_XL   = 3897648;    // 800*1024*4      = 3,276,800
static const size_t WS_XR   = 7174448;    // 800*1024*4
static const size_t WS_XA   = 10451248;   // 800*1024*4
static const size_t WS_XB   = 13728048;   // 800*1024*4      -> end ~17.0 MB

__device__ __forceinline__ float gelu_f(float x) {
    return 0.5f * x * (1.0f + erff(x * 0.70710678118654752440f));
}

// map float -> ordered uint (ascending) and back
__device__ __forceinline__ unsigned fkey(float f) {
    unsigned u = __float_as_uint(f);
    return (u & 0x80000000u) ? ~u : (u | 0x80000000u);
}
__device__ __forceinline__ float funkey(unsigned u) {
    unsigned v = (u & 0x80000000u) ? (u & 0x7FFFFFFFu) : ~u;
    return __uint_as_float(v);
}

// CDNA5 async copy: 16B global -> LDS, tracked with ASYNCcnt (per-lane addresses)
__device__ __forceinline__ void async_b128_to_lds(unsigned lds_off, const float* gaddr) {
    asm volatile("global_load_async_to_lds_b128 %0, %1, off"
                 :: "v"(lds_off), "v"(gaddr) : "memory");
}

// ---------------- 0: pad A to 16 rows (M=4 -> 16, zeros below) ----------------
__global__ void pad_a_kernel(const float* __restrict__ raw, float* __restrict__ Apad) {
    int idx = blockIdx.x * 256 + threadIdx.x;     // over 16*40000
    if (idx < 16 * NN) {
        int r = idx / NN;
        Apad[idx] = (r < 4) ? raw[idx] : 0.0f;
    }
}

// ---------------- 1: P = sigmoid(Apad @ Wg + bg), rows 0..3 kept ----------------
// 256 threads = 8 waves; block covers 128 columns. Wg is streamed through LDS in
// 32x128 fp32 chunks via global_load_async_to_lds_b128 (double-buffered, ASYNCcnt),
// WMMA B-fragments then come from LDS (row stride padded to 132 floats).
#define KCHUNK 32
#define BN     128
#define LDS_STRIDE 132

__global__ void __launch_bounds__(256) gemm_g_kernel(const float* __restrict__ Apad,
                                                     const float* __restrict__ Wg,
                                                     const float* __restrict__ bg,
                                                     float* __restrict__ P) {
    const int K = NN, N = NN;
    __shared__ __align__(16) float tile[2][KCHUNK][LDS_STRIDE];
    int t    = threadIdx.x;
    int wave = t >> 5;
    int lane = t & 31;
    int n_base  = blockIdx.x * BN;
    int tileIdx = blockIdx.x * 8 + wave;          // 16-col tile handled by this wave
    int m    = lane & 15;
    int half = lane >> 4;

    // loader mapping: thread loads 16B; rows i*8+wave, 4 floats starting at lane*4
    int gcol = n_base + lane * 4;
    if (gcol > N - 4) gcol = N - 4;               // clamp (last block tail unused)
    unsigned lbase0 = (unsigned)(size_t)(void*)&tile[0][0][0];
    unsigned lbase1 = (unsigned)(size_t)(void*)&tile[1][0][0];
    unsigned lane_off = (unsigned)(lane * 16);

    // prefetch chunk 0 into buffer 0
#pragma unroll
    for (int i = 0; i < 4; ++i) {
        int row = i * 8 + wave;
        async_b128_to_lds(lbase0 + (unsigned)(row * LDS_STRIDE * 4) + lane_off,
                          Wg + (size_t)row * N + gcol);
    }

    v8f c = {};
    const int nchunks = K / KCHUNK;               // 1250
    for (int ch = 0; ch < nchunks; ++ch) {
        int cur = ch & 1;
        int k0  = ch * KCHUNK;
        if (ch + 1 < nchunks) {
            unsigned lb = (cur ^ 1) ? lbase1 : lbase0;
#pragma unroll
            for (int i = 0; i < 4; ++i) {
                int row = i * 8 + wave;
                async_b128_to_lds(lb + (unsigned)(row * LDS_STRIDE * 4) + lane_off,
                                  Wg + (size_t)(k0 + KCHUNK + row) * N + gcol);
            }
            asm volatile("s_wait_asynccnt 0x4" ::: "memory");  // current chunk done
        } else {
            asm volatile("s_wait_asynccnt 0x0" ::: "memory");
        }
        __syncthreads();

        const float* a0 = Apad + (size_t)m * K + k0 + 2 * half;
        const float (*tb)[LDS_STRIDE] = tile[cur];
        int col = wave * 16 + m;
#pragma unroll
        for (int k = 0; k < KCHUNK; k += 4) {
            v2f a, b;
            a.x = a0[k];
            a.y = a0[k + 1];
            b.x = tb[k + 2 * half][col];
            b.y = tb[k + 1 + 2 * half][col];
            c = __builtin_amdgcn_wmma_f32_16x16x4_f32(false, a, false, b, (short)0, c, false, false);
        }
        __syncthreads();                          // protect buffer before next prefetch
    }

    if (tileIdx < 2500 && half == 0) {            // rows 0..3 live in c[0..3], lanes 0..15
        int n0 = tileIdx * 16;
        float bias = bg[n0 + m];
#pragma unroll
        for (int r = 0; r < 4; ++r) {
            float v = c[r] + bias;
            P[(size_t)r * N + n0 + m] = 1.0f / (1.0f + expf(-v));
        }
    }
}

// ---------------- 2: adj = 0.5*(a + a^T); row mean + unbiased std ----------------
__global__ void __launch_bounds__(256) sym_stats_kernel(const float* __restrict__ P,
                                                        float* __restrict__ adj,
                                                        float* __restrict__ base) {
    int b = blockIdx.x / NODES, i = blockIdx.x % NODES;
    __shared__ float row[NODES];
    __shared__ float red[256];
    const float* Pb = P + (size_t)b * NN;
    float s = 0.f;
    for (int j = threadIdx.x; j < NODES; j += 256) {
        float v = 0.5f * (Pb[i * NODES + j] + Pb[j * NODES + i]);
        row[j] = v;
        adj[(size_t)b * NN + i * NODES + j] = v;
        s += v;
    }
    red[threadIdx.x] = s;
    __syncthreads();
    for (int o = 128; o > 0; o >>= 1) {
        if (threadIdx.x < o) red[threadIdx.x] += red[threadIdx.x + o];
        __syncthreads();
    }
    float mean = red[0] / (float)NODES;
    __syncthreads();
    float s2 = 0.f;
    for (int j = threadIdx.x; j < NODES; j += 256) {
        float d = row[j] - mean;
        s2 += d * d;
    }
    red[threadIdx.x] = s2;
    __syncthreads();
    for (int o = 128; o > 0; o >>= 1) {
        if (threadIdx.x < o) red[threadIdx.x] += red[threadIdx.x + o];
        __syncthreads();
    }
    if (threadIdx.x == 0) {
        base[(b * NODES + i) * 2 + 0] = mean;
        base[(b * NODES + i) * 2 + 1] = sqrtf(red[0] / (float)(NODES - 1));
    }
}

// ---------------- 3: feature enhancer: MLP(2->8 GELU ->16) + LayerNorm ----------------
__global__ void enhancer_kernel(const float* __restrict__ base,
                                const float* __restrict__ W1, const float* __restrict__ b1,
                                const float* __restrict__ W2, const float* __restrict__ b2,
                                const float* __restrict__ g,  const float* __restrict__ be,
                                float* __restrict__ x0) {
    int node = blockIdx.x * 256 + threadIdx.x;
    if (node >= 800) return;
    float in0 = base[node * 2], in1 = base[node * 2 + 1];
    float h1[8], h[16];
#pragma unroll
    for (int j = 0; j < 8; ++j)
        h1[j] = gelu_f(in0 * W1[j] + in1 * W1[8 + j] + b1[j]);
#pragma unroll
    for (int cIt = 0; cIt < 16; ++cIt) {
        float acc = b2[cIt];
#pragma unroll
        for (int j = 0; j < 8; ++j) acc += h1[j] * W2[j * 16 + cIt];
        h[cIt] = acc;
    }
    float mu = 0.f;
#pragma unroll
    for (int cIt = 0; cIt < 16; ++cIt) mu += h[cIt];
    mu *= (1.0f / 16.0f);
    float var = 0.f;
#pragma unroll
    for (int cIt = 0; cIt < 16; ++cIt) { float d = h[cIt] - mu; var += d * d; }
    var *= (1.0f / 16.0f);
    float inv = rsqrtf(var + 1e-5f);
#pragma unroll
    for (int cIt = 0; cIt < 16; ++cIt)
        x0[node * 16 + cIt] = (h[cIt] - mu) * inv * g[cIt] + be[cIt];
}

// ---------------- 4: per-sample order statistics via 8-bit radix select ----------------
// blockIdx: b = blk>>1, k = 29999 + (blk&1)  (0-based ascending)
__global__ void __launch_bounds__(256) qselect_kernel(const float* __restrict__ adj,
                                                      float* __restrict__ sel) {
    int b = blockIdx.x >> 1;
    unsigned k = 29999u + (unsigned)(blockIdx.x & 1);
    const float* A = adj + (size_t)b * NN;
    __shared__ unsigned hist[256];
    __shared__ unsigned s_prefix, s_k;
    if (threadIdx.x == 0) { s_prefix = 0u; s_k = k; }
    __syncthreads();
    for (int pass = 0; pass < 4; ++pass) {
        int shift = 24 - pass * 8;
        hist[threadIdx.x] = 0u;
        __syncthreads();
        unsigned mask = (pass == 0) ? 0u : (0xFFFFFFFFu << (shift + 8));
        unsigned pref = s_prefix;
        for (int j = threadIdx.x; j < NN; j += 256) {
            unsigned u = fkey(A[j]);
            if ((u & mask) == pref) atomicAdd(&hist[(u >> shift) & 255u], 1u);
        }
        __syncthreads();
        if (threadIdx.x == 0) {
            unsigned kk = s_k, cum = 0u, bin = 0u;
            for (bin = 0u; bin < 256u; ++bin) {
                unsigned cth = hist[bin];
                if (cum + cth > kk) break;
                cum += cth;
            }
            s_k = kk - cum;
            s_prefix = s_prefix | (bin << shift);
        }
        __syncthreads();
    }
    if (threadIdx.x == 0) sel[b * 2 + (blockIdx.x & 1)] = funkey(s_prefix);
}

__global__ void thr_kernel(const float* __restrict__ sel, float* __restrict__ thr) {
    int b = threadIdx.x;
    if (b < 4) {
        float v1 = sel[b * 2], v2 = sel[b * 2 + 1];
        thr[b] = v1 + 0.25f * (v2 - v1);   // jnp.quantile linear interp at 0.75*(n-1)
    }
}

// ---------------- 5: generic C = A@W + bias via WMMA f32 16x16x4 ----------------
// A [M,K] row-major, W [K,N] row-major, M%16==0, N%16==0, K%4==0 (all L2-resident)
__global__ void __launch_bounds__(128) gemm_node_kernel(const float* __restrict__ A,
                                                        const float* __restrict__ W,
                                                        const float* __restrict__ bias,
                                                        float* __restrict__ Cout,
                                                        int M, int N, int K) {
    int wave = threadIdx.x >> 5;
    int lane = threadIdx.x & 31;
    int tilesN = N >> 4;
    int tilesTotal = (M >> 4) * tilesN;
    int tile = blockIdx.x * 4 + wave;
    if (tile >= tilesTotal) return;
    int mt = tile / tilesN, nt = tile - mt * tilesN;
    int m = lane & 15, half = lane >> 4;
    int n0 = nt * 16;

    v8f c = {};
    const float* a0 = A + (size_t)(mt * 16 + m) * K + 2 * half;
    const float* b0 = W + (size_t)(2 * half) * N + n0 + m;
#pragma unroll 4
    for (int k = 0; k < K; k += 4) {
        v2f a, b;
        a.x = a0[k];
        a.y = a0[k + 1];
        b.x = b0[(size_t)k * N];
        b.y = b0[(size_t)(k + 1) * N];
        c = __builtin_amdgcn_wmma_f32_16x16x4_f32(false, a, false, b, (short)0, c, false, false);
    }
    float bv = bias[n0 + m];
#pragma unroll
    for (int r = 0; r < 8; ++r) {
        int row = mt * 16 + r + half * 8;
        Cout[(size_t)row * N + n0 + m] = c[r] + bv;
    }
}

// ---------------- 6: dense masked GATv2 attention + aggregation + GELU ----------------
// One block per (b, dst). mask(s,d): selfloop ? (s==d || (adj>thr && s!=d)) : (adj>thr)
__global__ void __launch_bounds__(256) gat_attn_kernel(const float* __restrict__ xl,
                                                       const float* __restrict__ xr,
                                                       const float* __restrict__ att,
                                                       const float* __restrict__ bias,
                                                       const float* __restrict__ adj,
                                                       const float* __restrict__ thr,
                                                       float* __restrict__ out,
                                                       int H, int C, int selfloop) {
    int b = blockIdx.x / NODES, d = blockIdx.x % NODES;
    int HC = H * C;
    __shared__ float xrd[1024];
    __shared__ float sc[NODES * 8];
    __shared__ float hmax[8], hden[8];
    int dnode = b * NODES + d;

    for (int q = threadIdx.x; q < HC; q += 256)
        xrd[q] = xr[(size_t)dnode * HC + q];
    __syncthreads();

    float th = thr[b];
    const float* adjb = adj + (size_t)b * NN;
    int Pn = NODES * H;

    for (int p = threadIdx.x; p < Pn; p += 256) {
        int s = p / H, h = p - s * H;
        float a = adjb[s * NODES + d];   // row = src, col = dst
        bool edge = selfloop ? ((s == d) || (a > th && s != d)) : (a > th);
        float score = -__builtin_inff();
        if (edge) {
            const float* xls = xl + (size_t)(b * NODES + s) * HC + h * C;
            const float* ah  = att + h * C;
            const float* xrh = xrd + h * C;
            float acc = 0.f;
            for (int cc = 0; cc < C; ++cc) {
                float v = xls[cc] + xrh[cc];
                v = (v > 0.f) ? v : 0.2f * v;     // leaky_relu 0.2
                acc += ah[cc] * v;
            }
            score = acc;
        }
        sc[p] = score;
    }
    __syncthreads();

    if (threadIdx.x < (unsigned)H) {
        int h = threadIdx.x;
        float m = -__builtin_inff();
        for (int s = 0; s < NODES; ++s) m = fmaxf(m, sc[s * H + h]);
        if (!(m > -__builtin_inff())) m = 0.f;    // no valid in-edges
        float den = 0.f;
        for (int s = 0; s < NODES; ++s) den += expf(sc[s * H + h] - m);
        hmax[h] = m;
        hden[h] = den;
    }
    __syncthreads();

    for (int p = threadIdx.x; p < Pn; p += 256) {
        int h = p % H;
        float den = hden[h];
        sc[p] = (den > 0.f) ? expf(sc[p] - hmax[h]) / den : 0.f;  // exp(-inf)=0 kills masked
    }
    __syncthreads();

    for (int q = threadIdx.x; q < HC; q += 256) {
        int h = q / C;
        float acc = 0.f;
        const float* xlb = xl + (size_t)(b * NODES) * HC + q;
        for (int s = 0; s < NODES; ++s)
            acc += sc[s * H + h] * xlb[(size_t)s * HC];
        out[(size_t)dnode * HC + q] = gelu_f(acc + bias[q]);
    }
}

// ---------------- 7: global mean pool over nodes ----------------
__global__ void pool_kernel(const float* __restrict__ x, float* __restrict__ outp) {
    int b = blockIdx.x;
    int f = threadIdx.x;                          // 512 threads
    float acc = 0.f;
    for (int i = 0; i < NODES; ++i) acc += x[(size_t)(b * NODES + i) * 512 + f];
    outp[b * 512 + f] = acc * (1.0f / (float)NODES);
}

// ---------------- host launcher ----------------
extern "C" void kernel_launch(void* const* d_in, const int* in_sizes, int n_in,
                              void* d_out, int out_size, void* d_ws, size_t ws_size,
                              hipStream_t stream) {
    const float* raw_fc = (const float*)d_in[0];
    const float* Wg     = (const float*)d_in[1];
    const float* bg     = (const float*)d_in[2];
    const float* W1     = (const float*)d_in[3];
    const float* b1     = (const float*)d_in[4];
    const float* W2     = (const float*)d_in[5];
    const float* b2     = (const float*)d_in[6];
    const float* ln_g   = (const float*)d_in[7];
    const float* ln_b   = (const float*)d_in[8];
    const float* Wl[3]   = {(const float*)d_in[9],  (const float*)d_in[15], (const float*)d_in[21]};
    const float* bl[3]   = {(const float*)d_in[10], (const float*)d_in[16], (const float*)d_in[22]};
    const float* Wr[3]   = {(const float*)d_in[11], (const float*)d_in[17], (const float*)d_in[23]};
    const float* br[3]   = {(const float*)d_in[12], (const float*)d_in[18], (const float*)d_in[24]};
    const float* attw[3] = {(const float*)d_in[13], (const float*)d_in[19], (const float*)d_in[25]};
    const float* bsw[3]  = {(const float*)d_in[14], (const float*)d_in[20], (const float*)d_in[26]};

    char* ws = (char*)d_ws;
    float* Apad = (float*)(ws + WS_APAD);
    float* P    = (float*)(ws + WS_P);
    float* adj  = (float*)(ws + WS_ADJ);
    float* base = (float*)(ws + WS_BASE);
    float* sel  = (float*)(ws + WS_SEL);
    float* thr  = (float*)(ws + WS_THR);
    float* x0   = (float*)(ws + WS_X0);
    float* xl   = (float*)(ws + WS_XL);
    float* xr   = (float*)(ws + WS_XR);
    float* xA   = (float*)(ws + WS_XA);
    float* xB   = (float*)(ws + WS_XB);
    float* outp = (float*)d_out;

    // graph builder: sigmoid(raw_fc @ Wg + bg)  -- 6.4 GB stream, async-LDS + WMMA
    pad_a_kernel<<<(16 * NN + 255) / 256, 256, 0, stream>>>(raw_fc, Apad);
    gemm_g_kernel<<<313, 256, 0, stream>>>(Apad, Wg, bg, P);

    // symmetrize + node stats + enhancer
    sym_stats_kernel<<<800, 256, 0, stream>>>(P, adj, base);
    enhancer_kernel<<<4, 256, 0, stream>>>(base, W1, b1, W2, b2, ln_g, ln_b, x0);

    // exact 0.75 quantile per sample
    qselect_kernel<<<8, 256, 0, stream>>>(adj, sel);
    thr_kernel<<<1, 64, 0, stream>>>(sel, thr);

    // layer 1: in=16, H=8, C=128, no self loops
    {
        int M = 800, N = 1024, K = 16;
        int tiles = (M / 16) * (N / 16);
        gemm_node_kernel<<<(tiles + 3) / 4, 128, 0, stream>>>(x0, Wl[0], bl[0], xl, M, N, K);
        gemm_node_kernel<<<(tiles + 3) / 4, 128, 0, stream>>>(x0, Wr[0], br[0], xr, M, N, K);
        gat_attn_kernel<<<800, 256, 0, stream>>>(xl, xr, attw[0], bsw[0], adj, thr, xA, 8, 128, 0);
    }
    // layer 2: in=1024, H=4, C=256, self loops
    {
        int M = 800, N = 1024, K = 1024;
        int tiles = (M / 16) * (N / 16);
        gemm_node_kernel<<<(tiles + 3) / 4, 128, 0, stream>>>(xA, Wl[1], bl[1], xl, M, N, K);
        gemm_node_kernel<<<(tiles + 3) / 4, 128, 0, stream>>>(xA, Wr[1], br[1], xr, M, N, K);
        gat_attn_kernel<<<800, 256, 0, stream>>>(xl, xr, attw[1], bsw[1], adj, thr, xB, 4, 256, 1);
    }
    // layer 3: in=1024, H=1, C=512, self loops
    {
        int M = 800, N = 512, K = 1024;
        int tiles = (M / 16) * (N / 16);
        gemm_node_kernel<<<(tiles + 3) / 4, 128, 0, stream>>>(xB, Wl[2], bl[2], xl, M, N, K);
        gemm_node_kernel<<<(tiles + 3) / 4, 128, 0, stream>>>(xB, Wr[2], br[2], xr, M, N, K);
        gat_attn_kernel<<<800, 256, 0, stream>>>(xl, xr, attw[2], bsw[2], adj, thr, xA, 1, 512, 1);
    }
    // global mean pool -> [4, 512]
    pool_kernel<<<4, 512, 0, stream>>>(xA, outp);

    (void)in_sizes; (void)n_in; (void)out_size; (void)ws_size;
}